// Easy3DModel_60258391163527
// MI455X (gfx1250) — compile-verified
//
#include <hip/hip_runtime.h>

// Problem constants from the reference.
#define Qn 32
#define Vn 50000
#define Fn 256
#define Cn 10

// One 64-thread (2 wave32) block per (q, c) output row.
// Each thread handles one float4 -> 64 * 4 = 256 = F elements per row.
__global__ __launch_bounds__(64) void e3d_gather_emb_kernel(
    const float* __restrict__ pe,      // [Q, V, F] f32
    const int*   __restrict__ ids,     // [Q, C] i32
    const int*   __restrict__ labels,  // [Q, C] i32
    const float* __restrict__ emb,     // [3, F] f32
    float*       __restrict__ out)     // [Q, C, F] f32
{
    // 3 x 256 f32 interaction table staged in LDS (3 KB), 16B aligned for b128.
    __shared__ __align__(16) float s_emb[3 * Fn];

    const int t = threadIdx.x; // 0..63

    // ---- CDNA5 async global->LDS staging of the embedding table ----------
    // 3 KB = 192 x 16B chunks; 64 lanes x 3 iterations of b128 async copies.
    // Generic address of a __shared__ object has the LDS byte offset in its
    // low 32 bits on gfx1250, which is exactly what VDST of the async op wants.
    {
        const unsigned lds_base = (unsigned)(uintptr_t)(void*)s_emb;
#pragma unroll
        for (int i = 0; i < 3; ++i) {
            const int      chunk = t + i * 64;       // 0..191
            const float*   g     = emb + chunk * 4;  // 16B source
            const unsigned l     = lds_base + chunk * 16;
            asm volatile("global_load_async_to_lds_b128 %0, %1, off"
                         :
                         : "v"(l), "v"(g)
                         : "memory");
        }
        // Drain this wave's ASYNCcnt, then barrier so both waves see the LDS.
        asm volatile("s_wait_asynccnt 0" ::: "memory");
        __syncthreads();
    }

    // ---- Per-row gather + add --------------------------------------------
    const int b   = blockIdx.x;   // 0 .. Q*C-1
    const int q   = b / Cn;
    const int lab = labels[b];    // block-uniform -> scalarizable
    const int id  = ids[b];       // block-uniform -> scalarizable

    // Embedding row from LDS (lab in [0,3)).
    const float4* e4 = (const float4*)(s_emb + lab * Fn);
    float4 v = e4[t];

    // label == 2 means "no positional encoding": reference multiplies the
    // gathered row by 0, so we skip the 1 KB read entirely (uniform branch).
    if (lab != 2) {
        const float4* s4 =
            (const float4*)(pe + ((size_t)q * Vn + (size_t)id) * (size_t)Fn);
        const float4 a = s4[t];
        v.x += a.x;
        v.y += a.y;
        v.z += a.z;
        v.w += a.w;
    }

    float4* d4 = (float4*)(out + (size_t)b * (size_t)Fn);
    d4[t] = v;
}

extern "C" void kernel_launch(void* const* d_in, const int* in_sizes, int n_in,
                              void* d_out, int out_size, void* d_ws, size_t ws_size,
                              hipStream_t stream) {
    (void)in_sizes; (void)n_in; (void)out_size; (void)d_ws; (void)ws_size;

    const float* pe     = (const float*)d_in[0]; // [Q,V,F] f32
    const int*   ids    = (const int*)d_in[1];   // [Q,C] i32
    const int*   labels = (const int*)d_in[2];   // [Q,C] i32
    const float* emb    = (const float*)d_in[3]; // [3,F] f32
    float*       out    = (float*)d_out;         // [Q,C,F] f32

    e3d_gather_emb_kernel<<<dim3(Qn * Cn), dim3(64), 0, stream>>>(
        pe, ids, labels, emb, out);
}